// Autoencoder_65481071396632
// MI455X (gfx1250) — compile-verified
//
#include <hip/hip_runtime.h>
#include <math.h>

typedef float v2f __attribute__((ext_vector_type(2)));
typedef float v8f __attribute__((ext_vector_type(8)));

// One wave (32 threads) handles 16 samples. 4096 samples -> 256 blocks.
// z = [cos(x11), cos(x0)*cos(x11)]  (analytic collapse of the quantum circuit;
// ansatz_params provably do not affect the output).
// MLP 2->32->16->12 done with V_WMMA_F32_16X16X4_F32 in exact fp32.
__global__ __launch_bounds__(32) void qae_fused(
    const float* __restrict__ x,
    const float* __restrict__ W1, const float* __restrict__ b1,
    const float* __restrict__ W2, const float* __restrict__ b2,
    const float* __restrict__ W3, const float* __restrict__ b3,
    float* __restrict__ out)
{
    __shared__ float lds[768];            // [0..511]: h1 (16x32), [512..767]: h2 (16x16)
    const int  lane = threadIdx.x;        // 0..31 (wave32)
    const int  m    = lane & 15;          // A-row / B-col within tile
    const bool hi   = lane >= 16;         // upper half-wave -> K pair {2,3}
    const int  koff = hi ? 2 : 0;
    const int  b0   = blockIdx.x * 16;    // first sample of this tile

    // ---------------- encode: z0 = cos(x11), z1 = cos(x0)*cos(x11) ----------
    // A-matrix 16x4 fp32: lanes 0-15 hold {K=0,K=1}, lanes 16-31 hold {K=2,K=3}.
    // K=2,3 are zero padding (true K is 2).
    v2f a; a.x = 0.f; a.y = 0.f;
    if (!hi) {
        const float* xr = x + (size_t)(b0 + m) * 12;
        float c11 = cosf(xr[11]);
        float c0  = cosf(xr[0]);
        a.x = c11;        // K=0
        a.y = c0 * c11;   // K=1
    }

    // ---------------- layer 1: relu(z @ W1^T + b1) -> 16x32 ------------------
    // Two 16x16 output tiles; B tile = W1^T (2x32) with K padded to 4.
    #pragma unroll
    for (int t = 0; t < 2; ++t) {
        int o = t * 16 + m;                       // output column (0..31)
        v2f bm;
        bm.x = hi ? 0.f : W1[o * 2 + 0];          // B[K=0][N=o]
        bm.y = hi ? 0.f : W1[o * 2 + 1];          // B[K=1][N=o]
        float bias = b1[o];
        v8f c;
        #pragma unroll
        for (int v = 0; v < 8; ++v) c[v] = bias;  // same bias for every row M
        v8f d = __builtin_amdgcn_wmma_f32_16x16x4_f32(
                    false, a, false, bm, (short)0, c, false, false);
        #pragma unroll
        for (int v = 0; v < 8; ++v) {             // D layout: M=v(+8), N=lane&15
            int row = v + (hi ? 8 : 0);
            lds[row * 32 + t * 16 + m] = fmaxf(d[v], 0.f);
        }
    }
    __syncthreads();

    // ---------------- layer 2: relu(h1 @ W2^T + b2) -> 16x16 -----------------
    v8f c2;
    {
        float bias = b2[m];
        #pragma unroll
        for (int v = 0; v < 8; ++v) c2[v] = bias;
    }
    #pragma unroll
    for (int kb = 0; kb < 8; ++kb) {              // K = 32 -> 8 steps of 4
        int k = kb * 4 + koff;
        v2f a2;                                   // A from h1 in LDS
        a2.x = lds[m * 32 + k];
        a2.y = lds[m * 32 + k + 1];
        v2f bm;                                   // B[k][n] = W2[n][k]
        bm.x = W2[m * 32 + k];
        bm.y = W2[m * 32 + k + 1];
        c2 = __builtin_amdgcn_wmma_f32_16x16x4_f32(
                 false, a2, false, bm, (short)0, c2, false, false);
    }
    #pragma unroll
    for (int v = 0; v < 8; ++v) {
        int row = v + (hi ? 8 : 0);
        lds[512 + row * 16 + m] = fmaxf(c2[v], 0.f);
    }
    __syncthreads();

    // ---------------- layer 3: h2 @ W3^T + b3 -> 16x12 (N padded to 16) ------
    const bool nvalid = m < 12;
    v8f c3;
    {
        float bias = nvalid ? b3[m] : 0.f;
        #pragma unroll
        for (int v = 0; v < 8; ++v) c3[v] = bias;
    }
    #pragma unroll
    for (int kb = 0; kb < 4; ++kb) {              // K = 16 -> 4 steps of 4
        int k = kb * 4 + koff;
        v2f a3;
        a3.x = lds[512 + m * 16 + k];
        a3.y = lds[512 + m * 16 + k + 1];
        v2f bm;                                   // B[k][n] = W3[n][k], n<12
        bm.x = nvalid ? W3[m * 16 + k]     : 0.f;
        bm.y = nvalid ? W3[m * 16 + k + 1] : 0.f;
        c3 = __builtin_amdgcn_wmma_f32_16x16x4_f32(
                 false, a3, false, bm, (short)0, c3, false, false);
    }
    #pragma unroll
    for (int v = 0; v < 8; ++v) {
        int row = v + (hi ? 8 : 0);
        if (nvalid) out[(size_t)(b0 + row) * 12 + m] = c3[v];
    }
}

extern "C" void kernel_launch(void* const* d_in, const int* in_sizes, int n_in,
                              void* d_out, int out_size, void* d_ws, size_t ws_size,
                              hipStream_t stream) {
    (void)in_sizes; (void)n_in; (void)d_ws; (void)ws_size; (void)out_size;
    const float* x  = (const float*)d_in[0];
    // d_in[1] = ansatz_params: provably unused (CNOT+RZ tail is measurement-invariant)
    const float* W1 = (const float*)d_in[2];
    const float* b1 = (const float*)d_in[3];
    const float* W2 = (const float*)d_in[4];
    const float* b2 = (const float*)d_in[5];
    const float* W3 = (const float*)d_in[6];
    const float* b3 = (const float*)d_in[7];
    float* out = (float*)d_out;

    // 4096 samples / 16 per wave = 256 blocks of one wave each.
    qae_fused<<<256, 32, 0, stream>>>(x, W1, b1, W2, b2, W3, b3, out);
}